// GCNEncoder_57071525429814
// MI455X (gfx1250) — compile-verified
//
#include <hip/hip_runtime.h>
#include <stdint.h>

typedef float v2f __attribute__((ext_vector_type(2)));
typedef float v8f __attribute__((ext_vector_type(8)));

#define DHID 128   // hidden width (also concat width of [W_mu|W_ls])
#define BROW 136   // LDS row stride for B tile (padding -> conflict-free dual-row reads)

// ---------------- degree / norm ----------------
__global__ void init_deg(float* deg, int N) {
    int i = blockIdx.x * blockDim.x + threadIdx.x;
    if (i < N) deg[i] = 1.0f;                       // self-loop contributes 1
}

__global__ void count_deg(const long long* __restrict__ dst, float* deg, int E) {
    int i = blockIdx.x * blockDim.x + threadIdx.x;
    if (i < E) atomicAdd(deg + (int)dst[i], 1.0f);
}

__global__ void rsqrt_deg(float* deg, int N) {
    int i = blockIdx.x * blockDim.x + threadIdx.x;
    if (i < N) deg[i] = rsqrtf(deg[i]);             // deg >= 1 (self loop), no zero check
}

// ---------------- WMMA fp32 GEMM: C[N x 128] = A[N x 128] @ B[128 x 128] ----------------
// block = 256 threads = 8 waves; each wave owns one 16x16 output tile (wave w -> cols 16w).
// B (64KB) is staged once per block into LDS with a padded row stride so the two
// lane-halves (rows k and k+2: 272 floats apart = +16 banks) hit disjoint bank windows.
__global__ __launch_bounds__(256) void wmma_gemm128(const float* __restrict__ A,
                                                    const float* __restrict__ B,
                                                    float* __restrict__ C, int Nrows) {
    __shared__ float Bs[DHID * BROW];

    // cooperative stage: 128 rows x 32 float4 per row
    {
        const float4* Bv = (const float4*)B;
        for (int i = threadIdx.x; i < DHID * 32; i += 256) {
            int r = i >> 5, c4 = i & 31;
            *(float4*)(&Bs[r * BROW + c4 * 4]) = Bv[i];
        }
    }
    __syncthreads();

    const int lane = threadIdx.x & 31;
    const int wave = threadIdx.x >> 5;
    const int row0 = blockIdx.x << 4;
    const int col0 = wave << 4;
    const int m    = lane & 15;
    const int kh   = (lane >> 4) << 1;   // 0 for lanes 0-15, 2 for lanes 16-31

    // clamp A row so EXEC stays all-ones through the WMMA loop (ISA requirement)
    int arow_i = row0 + m; if (arow_i >= Nrows) arow_i = Nrows - 1;
    const float* arow = A + (size_t)arow_i * DHID;

    v8f acc = {0.f,0.f,0.f,0.f,0.f,0.f,0.f,0.f};
    #pragma unroll 4
    for (int k = 0; k < DHID; k += 4) {
        // A 16x4: lane half selects K pair {kh, kh+1}; contiguous -> one b64 load
        v2f a = *(const v2f*)(arow + k + kh);
        // B 4x16 from LDS: VGPR0 row K=kh, VGPR1 row K=kh+1, col = lane&15
        v2f b;
        b.x = Bs[(k + kh)     * BROW + col0 + m];
        b.y = Bs[(k + kh + 1) * BROW + col0 + m];
        acc = __builtin_amdgcn_wmma_f32_16x16x4_f32(false, a, false, b,
                                                    (short)0, acc, false, false);
    }

    // C/D layout: VGPR v -> row M = v + 8*(lane>=16), col = lane&15
    const int cr = (lane >> 4) << 3;
    if (row0 + 16 <= Nrows) {             // tile-uniform fast path: no divergence
        float* crow = C + (size_t)(row0 + cr) * DHID + col0 + m;
        #pragma unroll
        for (int v = 0; v < 8; ++v) crow[(size_t)v * DHID] = acc[v];
    } else {
        #pragma unroll
        for (int v = 0; v < 8; ++v) {
            int r = row0 + cr + v;
            if (r < Nrows) C[(size_t)r * DHID + col0 + m] = acc[v];
        }
    }
}

// ---------------- edge aggregation: out[dst] += msg[src] * dinv[src]*dinv[dst] ----------------
// one wave per edge; lane handles 4 contiguous columns (float4). Edges e >= E are the
// implicit self-loops (src = dst = e - E). Same-address lane loads of src/dst broadcast in L1.
__global__ __launch_bounds__(256) void scatter_add(const float* __restrict__ msg,
                                                   const long long* __restrict__ srcI,
                                                   const long long* __restrict__ dstI,
                                                   const float* __restrict__ dinv,
                                                   float* __restrict__ out, int E, int N) {
    int gw   = (int)(((unsigned)blockIdx.x * blockDim.x + threadIdx.x) >> 5);
    int lane = threadIdx.x & 31;
    int total = E + N;
    if (gw >= total) return;
    long long s, d;
    if (gw < E) { s = srcI[gw]; d = dstI[gw]; }
    else        { s = d = (long long)(gw - E); }
    float nrm = dinv[(int)s] * dinv[(int)d];
    float4 v = *(const float4*)(msg + (size_t)s * DHID + lane * 4);
    float* orow = out + (size_t)d * DHID + lane * 4;
    atomicAdd(orow + 0, v.x * nrm);
    atomicAdd(orow + 1, v.y * nrm);
    atomicAdd(orow + 2, v.z * nrm);
    atomicAdd(orow + 3, v.w * nrm);
}

// ---------------- elementwise (float4-vectorized) ----------------
__global__ void bias_relu(float* __restrict__ h, const float* __restrict__ b, int total4) {
    int i = blockIdx.x * blockDim.x + threadIdx.x;   // total4 = N*128/4
    if (i < total4) {
        float4 v = ((float4*)h)[i];
        float4 bb = ((const float4*)b)[i & 31];      // 128 cols = 32 float4s
        v.x += bb.x; v.y += bb.y; v.z += bb.z; v.w += bb.w;
        v.x = v.x > 0.f ? v.x : 0.f;
        v.y = v.y > 0.f ? v.y : 0.f;
        v.z = v.z > 0.f ? v.z : 0.f;
        v.w = v.w > 0.f ? v.w : 0.f;
        ((float4*)h)[i] = v;
    }
}

__global__ void pack_w(const float* __restrict__ Wmu, const float* __restrict__ Wls,
                       float* __restrict__ Wcat) {
    int i = blockIdx.x * blockDim.x + threadIdx.x;   // 128*128 threads
    int r = i >> 7, c = i & 127;
    Wcat[i] = (c < 64) ? Wmu[r * 64 + c] : Wls[r * 64 + (c - 64)];
}

// split aggregated [N x 128] into mu (cols 0-63) and logstd (cols 64-127), add biases
__global__ void finalize_out(const float* __restrict__ agg, const float* __restrict__ bmu,
                             const float* __restrict__ bls, float* __restrict__ out, int N) {
    int i = blockIdx.x * blockDim.x + threadIdx.x;   // N*16 float4s per output half
    int total4 = N * 16;
    if (i < total4) {
        int row = i >> 4, c4 = i & 15;               // 64 cols = 16 float4s
        const float4* arow = (const float4*)(agg + (size_t)row * DHID);
        float4 vmu = arow[c4],      bmu4 = ((const float4*)bmu)[c4];
        float4 vls = arow[16 + c4], bls4 = ((const float4*)bls)[c4];
        vmu.x += bmu4.x; vmu.y += bmu4.y; vmu.z += bmu4.z; vmu.w += bmu4.w;
        vls.x += bls4.x; vls.y += bls4.y; vls.z += bls4.z; vls.w += bls4.w;
        ((float4*)out)[i]          = vmu;
        ((float4*)out)[total4 + i] = vls;
    }
}

extern "C" void kernel_launch(void* const* d_in, const int* in_sizes, int n_in,
                              void* d_out, int out_size, void* d_ws, size_t ws_size,
                              hipStream_t stream) {
    const float*     x    = (const float*)d_in[0];
    const long long* eidx = (const long long*)d_in[1];   // [2, E] int64
    const float*     W1   = (const float*)d_in[2];
    const float*     b1   = (const float*)d_in[3];
    const float*     Wmu  = (const float*)d_in[4];
    const float*     bmu  = (const float*)d_in[5];
    const float*     Wls  = (const float*)d_in[6];
    const float*     bls  = (const float*)d_in[7];
    float*           out  = (float*)d_out;

    const int N = in_sizes[0] / DHID;        // 100000
    const int E = in_sizes[1] / 2;           // 1600000
    const long long* srcI = eidx;
    const long long* dstI = eidx + E;

    // ws layout: dinv[N] | bufA[N*128] | bufB[N*128] | Wcat[128*128]
    float* dinv = (float*)d_ws;
    float* bufA = dinv + N;
    float* bufB = bufA + (size_t)N * DHID;
    float* Wcat = bufB + (size_t)N * DHID;

    const int T = 256;
    const int waves = E + N;                           // one wave per edge (incl self-loops)
    const int sblocks = (waves * 32 + T - 1) / T;
    const int rtiles = (N + 15) / 16;

    // --- normalization coefficients ---
    init_deg <<<(N + T - 1) / T, T, 0, stream>>>(dinv, N);
    count_deg<<<(E + T - 1) / T, T, 0, stream>>>(dstI, dinv, E);
    rsqrt_deg<<<(N + T - 1) / T, T, 0, stream>>>(dinv, N);

    // --- layer 1: h = relu( scatter( (x@W1)[src]*norm ) + b1 ) ---
    wmma_gemm128<<<rtiles, T, 0, stream>>>(x, W1, bufA, N);
    hipMemsetAsync(bufB, 0, (size_t)N * DHID * sizeof(float), stream);
    scatter_add <<<sblocks, T, 0, stream>>>(bufA, srcI, dstI, dinv, bufB, E, N);
    bias_relu   <<<(N * 32 + T - 1) / T, T, 0, stream>>>(bufB, b1, N * 32);

    // --- layer 2 (fused mu & logstd): z = h @ [W_mu|W_ls], aggregate once ---
    pack_w      <<<(DHID * DHID) / T, T, 0, stream>>>(Wmu, Wls, Wcat);
    wmma_gemm128<<<rtiles, T, 0, stream>>>(bufB, Wcat, bufA, N);
    hipMemsetAsync(bufB, 0, (size_t)N * DHID * sizeof(float), stream);
    scatter_add <<<sblocks, T, 0, stream>>>(bufA, srcI, dstI, dinv, bufB, E, N);

    finalize_out<<<(N * 16 + T - 1) / T, T, 0, stream>>>(bufB, bmu, bls, out, N);
}